// WildcatPool2d_42880953483991
// MI455X (gfx1250) — compile-verified
//
#include <hip/hip_runtime.h>
#include <stdint.h>

// Problem geometry (reference: x[32, 64, 64, 512] fp32, k=20 both ends)
#define NROWS   4096      // H*W
#define NCH     512       // C
#define CBLK    8         // channels per workgroup (= waves per workgroup)
#define RSTRIDE 9         // LDS row stride in floats (8 data + 1 pad dword from TDM pad)
#define KSEL    20
#define CAP     256       // candidate buffer capacity per (channel, end)

// LDS layout (dynamic):
//   [0, 147456)           : float data[4096][9]  (TDM writes 8 floats + 1 pad per row)
//   [147456, 147520)      : unsigned cnts[16]    (per-wave top/bot candidate counters)
//   [147520, 163904)      : float bufs[8][2][CAP]
#define SMEM_DATA_BYTES (NROWS * RSTRIDE * 4)
#define SMEM_CNT_BYTES  64
#define SMEM_BUF_BYTES  (CBLK * 2 * CAP * 4)
#define SMEM_TOTAL      (SMEM_DATA_BYTES + SMEM_CNT_BYTES + SMEM_BUF_BYTES)

typedef __attribute__((ext_vector_type(4))) unsigned int v4u;
typedef __attribute__((ext_vector_type(8))) int v8i;
typedef __attribute__((ext_vector_type(4))) int v4i;

// Monotone order-preserving float <-> uint transform
__device__ __forceinline__ unsigned ordkey(float f) {
  unsigned u = __float_as_uint(f);
  return u ^ ((unsigned)((int)u >> 31) | 0x80000000u);
}
__device__ __forceinline__ float ordinv(unsigned k) {
  unsigned u = (k & 0x80000000u) ? (k ^ 0x80000000u) : ~k;
  return __uint_as_float(u);
}

// wave32 reductions via lane shuffles
__device__ __forceinline__ unsigned wred_maxu(unsigned v) {
#pragma unroll
  for (int off = 16; off > 0; off >>= 1) {
    unsigned o = (unsigned)__shfl_xor((int)v, off, 32);
    v = (o > v) ? o : v;
  }
  return v;
}
__device__ __forceinline__ unsigned wred_minu(unsigned v) {
#pragma unroll
  for (int off = 16; off > 0; off >>= 1) {
    unsigned o = (unsigned)__shfl_xor((int)v, off, 32);
    v = (o < v) ? o : v;
  }
  return v;
}
__device__ __forceinline__ int wred_addi(int v) {
#pragma unroll
  for (int off = 16; off > 0; off >>= 1) v += __shfl_xor(v, off, 32);
  return v;
}

// Exact sum of the KSEL most-extreme values among buf[0..n), 20 <= n <= CAP.
// top=true: largest; top=false: smallest. Tie-exact via multiplicity counting.
__device__ float select_from_buf(const float* buf, int n, bool top) {
  const int lane = threadIdx.x & 31;
  unsigned key[CAP / 32];
#pragma unroll
  for (int s = 0; s < CAP / 32; ++s) {
    int idx = lane + s * 32;
    unsigned k = 0u;
    if (idx < n) {
      k = ordkey(buf[idx]);
      if (!top) k = ~k;          // unify: always reduce with max, sentinel 0
    }
    key[s] = k;
  }
  float sum = 0.0f;
  int remaining = KSEL;
  for (int r = 0; r < KSEL && remaining > 0; ++r) {
    unsigned m = 0u;
#pragma unroll
    for (int s = 0; s < CAP / 32; ++s) m = (key[s] > m) ? key[s] : m;
    m = wred_maxu(m);
    if (m == 0u) break;
    int c = 0;
#pragma unroll
    for (int s = 0; s < CAP / 32; ++s) c += (key[s] == m) ? 1 : 0;
    c = wred_addi(c);
    int take = (c < remaining) ? c : remaining;
    sum += ordinv(top ? m : ~m) * (float)take;
    remaining -= take;
#pragma unroll
    for (int s = 0; s < CAP / 32; ++s) if (key[s] == m) key[s] = 0u;
  }
  return sum;
}

// Rare exact fallback: 20 rounds of wave-wide bounded-max over all 4096 values in LDS.
__device__ float select_serial(const float* data, int wv, bool top) {
  const int lane = threadIdx.x & 31;
  unsigned bound = 0xFFFFFFFFu;   // exclusive upper bound on transformed key
  float sum = 0.0f;
  int remaining = KSEL;
  for (int r = 0; r < KSEL && remaining > 0; ++r) {
    unsigned mloc = 0u;
    for (int i = 0; i < NROWS / 32; ++i) {
      unsigned k = ordkey(data[(i * 32 + lane) * RSTRIDE + wv]);
      if (!top) k = ~k;
      if (k < bound && k > mloc) mloc = k;
    }
    unsigned m = wred_maxu(mloc);
    if (m == 0u) break;
    int c = 0;
    for (int i = 0; i < NROWS / 32; ++i) {
      unsigned k = ordkey(data[(i * 32 + lane) * RSTRIDE + wv]);
      if (!top) k = ~k;
      c += (k == m) ? 1 : 0;
    }
    c = wred_addi(c);
    int take = (c < remaining) ? c : remaining;
    sum += ordinv(top ? m : ~m) * (float)take;
    remaining -= take;
    bound = m;
  }
  return sum;
}

extern __shared__ char smem_raw[];

__global__ __launch_bounds__(256) void wildcat_pool_kernel(
    const float* __restrict__ x, float* __restrict__ out) {
  float*    data = (float*)smem_raw;
  unsigned* cnts = (unsigned*)(smem_raw + SMEM_DATA_BYTES);
  float*    bufs = (float*)(smem_raw + SMEM_DATA_BYTES + SMEM_CNT_BYTES);

  const int lane = threadIdx.x & 31;
  const int wv   = threadIdx.x >> 5;        // wave id == channel offset 0..7
  const int b    = blockIdx.x >> 6;         // 64 channel-blocks per batch
  const int c0   = (blockIdx.x & 63) * CBLK;

  // ---- TDM: each wave DMAs its 512-row slice of the [4096 x 8] fp32 tile into LDS.
  // pad_enable: +1 dword after every 8 dwords -> LDS rows land at stride 9 floats,
  // which makes the later column reads (stride 9, coprime with 64 banks) conflict-free.
  {
    const int row0 = wv * (NROWS / CBLK);
    uint64_t ga = (uint64_t)(uintptr_t)x +
                  (((uint64_t)(b * NROWS + row0) * (uint64_t)NCH + (uint64_t)c0) * 4u);
    unsigned lds = (unsigned)(uintptr_t)data + (unsigned)(row0 * RSTRIDE * 4);
    // D# group 0: count=1 | lds_addr | global_addr lo | global_addr hi + type=2
    v4u g0 = { 1u, lds, (unsigned)(ga & 0xFFFFFFFFu),
               (unsigned)((ga >> 32) & 0x1FFFFFFu) | 0x80000000u };
    // D# group 1:
    //  w0: data_size=2 (4B) | pad_enable | pad_interval=2 (8 dwords) | pad_amount=0 (1 dword)
    //  w1[31:16]=tensor_dim0 lo16 (512)   w2[31:16]=tensor_dim1 lo16 (512 rows/slice)
    //  w3[31:16]=tile_dim0 (8)            w4[15:0]=tile_dim1 (512)
    //  w5=tensor_dim0_stride (512 elems)  w6,w7=0 (2D)
    v8i g1 = { (int)((2u << 16) | (1u << 20) | (2u << 22)),
               (int)(512u << 16),
               (int)(512u << 16),
               (int)(8u << 16),
               (int)512, (int)512, 0, 0 };
    v4i gz4 = { 0, 0, 0, 0 };
    v8i gz8 = { 0, 0, 0, 0, 0, 0, 0, 0 };
    __builtin_amdgcn_tensor_load_to_lds(g0, g1, gz4, gz4, gz8, 0);
    __builtin_amdgcn_s_wait_tensorcnt(0);
  }
  __syncthreads();

  if (lane == 0) { cnts[wv * 2] = 0u; cnts[wv * 2 + 1] = 0u; }

  // ---- 256 evenly spaced samples -> pivots ~ 8th largest / 8th smallest sample
  unsigned ks[8];
#pragma unroll
  for (int j = 0; j < 8; ++j) {
    int row = (lane + j * 32) * 16;                 // covers rows 0..4080
    ks[j] = ordkey(data[row * RSTRIDE + wv]);
  }
  unsigned t[8], pivT = 0u, pivB = 0xFFFFFFFFu;
#pragma unroll
  for (int j = 0; j < 8; ++j) t[j] = ks[j];
  for (int r = 0; r < 8; ++r) {                     // 8th-largest-ish (dups collapse -> safe)
    unsigned m = 0u;
#pragma unroll
    for (int j = 0; j < 8; ++j) m = (t[j] > m) ? t[j] : m;
    m = wred_maxu(m);
    pivT = m;
#pragma unroll
    for (int j = 0; j < 8; ++j) if (t[j] == m) t[j] = 0u;
  }
#pragma unroll
  for (int j = 0; j < 8; ++j) t[j] = ks[j];
  for (int r = 0; r < 8; ++r) {                     // 8th-smallest-ish
    unsigned m = 0xFFFFFFFFu;
#pragma unroll
    for (int j = 0; j < 8; ++j) m = (t[j] < m) ? t[j] : m;
    m = wred_minu(m);
    pivB = m;
#pragma unroll
    for (int j = 0; j < 8; ++j) if (t[j] == m) t[j] = 0xFFFFFFFFu;
  }

  // ---- compaction pass: 1 DS load + 2 compares per element; appends are rare
  float* bufT = bufs + wv * (2 * CAP);
  float* bufB = bufT + CAP;
  unsigned* cT = &cnts[wv * 2];
  unsigned* cB = cT + 1;
  for (int i = 0; i < NROWS / 32; ++i) {
    float v = data[(i * 32 + lane) * RSTRIDE + wv];
    unsigned u = ordkey(v);
    if (u > pivT) { unsigned p = atomicAdd(cT, 1u); if (p < CAP) bufT[p] = v; }
    if (u < pivB) { unsigned p = atomicAdd(cB, 1u); if (p < CAP) bufB[p] = v; }
  }
  __syncthreads();
  unsigned nT = *cT, nB = *cB;

  float sumT = (nT >= KSEL && nT <= CAP) ? select_from_buf(bufT, (int)nT, true)
                                         : select_serial(data, wv, true);
  float sumB = (nB >= KSEL && nB <= CAP) ? select_from_buf(bufB, (int)nB, false)
                                         : select_serial(data, wv, false);

  if (lane == 0) out[b * NCH + c0 + wv] = (sumT + sumB) * (1.0f / (float)KSEL);
}

extern "C" void kernel_launch(void* const* d_in, const int* in_sizes, int n_in,
                              void* d_out, int out_size, void* d_ws, size_t ws_size,
                              hipStream_t stream) {
  (void)in_sizes; (void)n_in; (void)d_ws; (void)ws_size; (void)out_size;
  const float* x = (const float*)d_in[0];
  float* out = (float*)d_out;
  (void)hipFuncSetAttribute((const void*)wildcat_pool_kernel,
                            hipFuncAttributeMaxDynamicSharedMemorySize,
                            (int)SMEM_TOTAL);
  // 32 batches x 64 channel-blocks; 256 threads (8 wave32) per block
  wildcat_pool_kernel<<<dim3(32 * 64), dim3(256), SMEM_TOTAL, stream>>>(x, out);
}